// HistAttnBlock_89189290869364
// MI455X (gfx1250) — compile-verified
//
#include <hip/hip_runtime.h>
#include <math.h>

// ---------------------------------------------------------------------------
// HistAttnBlock for MI455X (gfx1250, wave32).
// hist (B,L,L,C,NB) staged in d_out (exactly 8M floats, dead after pass2).
// conv_transpose(stride2,k2) == GEMM M=32768,K=8,N=128 -> V_WMMA_F32_16X16X4_F32.
// ---------------------------------------------------------------------------

#define Bb 32
#define Cc 32
#define Hh 64
#define Ww 64
#define NB 8
#define LL 32
#define OCC 32
#define NBINS 8192      // B*C*NB
#define REG_ELEMS 9216  // B*C*3*3

typedef __attribute__((ext_vector_type(2))) float v2f;
typedef __attribute__((ext_vector_type(8))) float v8f;

__global__ __launch_bounds__(256) void k_init(float* acc) {
  int t = threadIdx.x;
  if (t < 192) acc[t] = 0.0f;
}

// ---- Pass 1: per-region histogram -> normalized hist + per-C BN stats ------
__global__ __launch_bounds__(256) void k_hist(const float* __restrict__ x,
                                              float* __restrict__ hist, // (B,L,L,C,NB)
                                              float* __restrict__ hsum,
                                              float* __restrict__ hsumsq) {
  const int r  = blockIdx.x;          // region id = li*32+lj
  const int li = r >> 5, lj = r & 31;
  const int tid = threadIdx.x;
  // _region_grid(): wl=3, wl2=0, b=61/31 (double, mirrors numpy exactly)
  const double bstep = (double)(Ww - 3) / (double)(LL - 1);
  const int r0 = (int)floor((double)li * bstep);
  const int c0 = (int)floor((double)lj * bstep);

  __shared__ float sh[NBINS];           // 32KB count histogram
  __shared__ float rmn[8], rmx[8];
  __shared__ float sMn, sMx;

  // zero histogram
  for (int i = tid; i < NBINS; i += 256) sh[i] = 0.0f;

  // gather region: 3072 row-segments of 3 floats; cache 36 values/thread
  float vv[36];
  float mn = 3.4e38f, mx = -3.4e38f;
  int q = 0;
  for (int s = tid; s < 3072; s += 256) {
    const int bc = s / 3, a = s - 3 * bc;       // bc = b*32+c
    const float* p = x + (size_t)bc * 4096 + (size_t)(r0 + a) * 64 + c0;
    #pragma unroll
    for (int d = 0; d < 3; ++d) {
      float v = p[d];
      vv[q + d] = v;
      mn = fminf(mn, v); mx = fmaxf(mx, v);
    }
    q += 3;
  }
  // wave32 reduce then cross-wave
  #pragma unroll
  for (int off = 16; off > 0; off >>= 1) {
    mn = fminf(mn, __shfl_xor(mn, off, 32));
    mx = fmaxf(mx, __shfl_xor(mx, off, 32));
  }
  if ((tid & 31) == 0) { rmn[tid >> 5] = mn; rmx[tid >> 5] = mx; }
  __syncthreads();
  if (tid == 0) {
    float a = rmn[0], b = rmx[0];
    #pragma unroll
    for (int i = 1; i < 8; ++i) { a = fminf(a, rmn[i]); b = fmaxf(b, rmx[i]); }
    sMn = a; sMx = b;
  }
  __syncthreads();
  mn = sMn;
  const float scale = (float)NBINS / fmaxf(sMx - mn, 1e-12f);

  // LDS float histogram (ds_add_f32)
  #pragma unroll
  for (int e = 0; e < 36; ++e) {
    int bin = (int)floorf((vv[e] - mn) * scale);
    bin = bin < 0 ? 0 : (bin > NBINS - 1 ? NBINS - 1 : bin);
    atomicAdd(&sh[bin], 1.0f);
  }
  __syncthreads();

  // normalize groups of NB=8 over (b,c); thread's channel c = tid&31 is fixed
  float csum = 0.0f, csq = 0.0f;
  for (int g = tid; g < 1024; g += 256) {     // g = b*32+c
    const int b = g >> 5, ch = g & 31;
    float sq = 0.0f;
    #pragma unroll
    for (int nb = 0; nb < NB; ++nb) { float v = sh[g * 8 + nb]; sq += v * v; }
    const float inv = 1.0f / (sqrtf(sq) + 1e-7f);
    float* o = hist + (size_t)b * 262144 + (size_t)r * 256 + (size_t)ch * 8;
    #pragma unroll
    for (int nb = 0; nb < NB; ++nb) {
      float h = sh[g * 8 + nb] * inv;
      o[nb] = h;
      csum += h; csq += h * h;
    }
  }
  __syncthreads();
  if (tid < 64) sh[tid] = 0.0f;
  __syncthreads();
  atomicAdd(&sh[tid & 31], csum);
  atomicAdd(&sh[32 + (tid & 31)], csq);
  __syncthreads();
  if (tid < 32) { atomicAdd(&hsum[tid], sh[tid]); atomicAdd(&hsumsq[tid], sh[32 + tid]); }
}

// ---- Pass 2: attn = h*(h*s_c + t_c); reduce over C and over NB -------------
__global__ __launch_bounds__(256) void k_attn(const float* __restrict__ hist,
                                              const float* __restrict__ g3d,
                                              const float* __restrict__ b3d,
                                              const float* __restrict__ hsum,
                                              const float* __restrict__ hsumsq,
                                              float* __restrict__ ar2,   // (B,L,L,NB)
                                              float* __restrict__ acs,   // (B,C,L,L)
                                              float* __restrict__ asum,
                                              float* __restrict__ asumsq) {
  const int tid = threadIdx.x;
  __shared__ float sc[32], tc[32], bs[32], bq[32];
  if (tid < 32) {
    const float inv_n = 1.0f / 262144.0f;   // B*NB*L*L
    float mean = hsum[tid] * inv_n;
    float var  = hsumsq[tid] * inv_n - mean * mean;
    float s = g3d[tid] * rsqrtf(var + 1e-5f);
    sc[tid] = s;
    tc[tid] = b3d[tid] - mean * s;
    bs[tid] = 0.0f; bq[tid] = 0.0f;
  }
  __syncthreads();

  const int p  = blockIdx.x * 256 + tid;   // position in [0, 32768)
  const int b  = p >> 10, sp = p & 1023;
  const float* hrow = hist + (size_t)b * 262144 + (size_t)sp * 256;

  float red[NB];
  #pragma unroll
  for (int nb = 0; nb < NB; ++nb) red[nb] = 0.0f;

  for (int c = 0; c < Cc; ++c) {
    const float s = sc[c], t = tc[c];
    float cs = 0.0f;
    #pragma unroll
    for (int nb = 0; nb < NB; ++nb) {
      float h = hrow[c * 8 + nb];
      float a = h * (h * s + t);
      red[nb] += a; cs += a;
    }
    acs[((size_t)(b * 32 + c) << 10) + sp] = cs;
    atomicAdd(&bs[c], cs);
    atomicAdd(&bq[c], cs * cs);
  }
  #pragma unroll
  for (int nb = 0; nb < NB; ++nb) ar2[(size_t)p * 8 + nb] = red[nb];
  __syncthreads();
  if (tid < 32) { atomicAdd(&asum[tid], bs[tid]); atomicAdd(&asumsq[tid], bq[tid]); }
}

// ---- Pass 3: conv_transpose as WMMA GEMM (M=32768, K=8, N=128) -------------
// A(16x4 f32): lanes 0-15 hold {K0,K1} of row M=lane; lanes 16-31 hold {K2,K3}.
// B(4x16 f32): lanes 0-15 hold {K0,K1} at col N=lane; lanes 16-31 {K2,K3}.
// D: VGPR v -> row v (lanes 0-15) / row v+8 (lanes 16-31), col = lane&15.
__global__ __launch_bounds__(256) void k_convt(const float* __restrict__ ar2,
                                               const float* __restrict__ wct,  // (8,32,2,2) flat 8x128
                                               const float* __restrict__ bct,
                                               float* __restrict__ up,         // d_out[0..4M)
                                               float* __restrict__ usum,
                                               float* __restrict__ usumsq) {
  const int tid  = threadIdx.x;
  const int lane = tid & 31, wv = tid >> 5;
  const int m0 = blockIdx.x * 16;        // 2048 blocks cover M=32768
  const int n0 = wv * 16;                // 8 waves cover N=128
  const int half = lane >> 4, l16 = lane & 15;
  const int kb = half * 2;

  const float* arow = ar2 + (size_t)(m0 + l16) * 8;
  v2f a0, a1, b0, b1;
  a0.x = arow[kb];     a0.y = arow[kb + 1];
  a1.x = arow[kb + 4]; a1.y = arow[kb + 5];
  const int n = n0 + l16;
  b0.x = wct[kb * 128 + n];       b0.y = wct[(kb + 1) * 128 + n];
  b1.x = wct[(kb + 4) * 128 + n]; b1.y = wct[(kb + 5) * 128 + n];

  v8f c = {0.f, 0.f, 0.f, 0.f, 0.f, 0.f, 0.f, 0.f};
  c = __builtin_amdgcn_wmma_f32_16x16x4_f32(false, a0, false, b0, (short)0, c, false, false);
  c = __builtin_amdgcn_wmma_f32_16x16x4_f32(false, a1, false, b1, (short)0, c, false, false);

  // scatter: n -> (oc, ki, kj); m -> (b, i, j); out[b,oc,2i+ki,2j+kj]
  const int oc = n >> 2, ki = (n >> 1) & 1, kj = n & 1;
  const float bias = bct[oc];
  float lsum = 0.0f, lsq = 0.0f;
  #pragma unroll
  for (int v = 0; v < 8; ++v) {
    const int M = m0 + v + half * 8;
    const int bI = M >> 10, sp = M & 1023, i = sp >> 5, j = sp & 31;
    const float val = c[v] + bias;
    up[(((size_t)(bI * 32 + oc)) << 12) + (size_t)(2 * i + ki) * 64 + (2 * j + kj)] = val;
    lsum += val; lsq += val * val;
  }
  __shared__ float bs[32], bq[32];
  if (tid < 32) { bs[tid] = 0.0f; bq[tid] = 0.0f; }
  __syncthreads();
  atomicAdd(&bs[oc], lsum);
  atomicAdd(&bq[oc], lsq);
  __syncthreads();
  if (tid < 32) { atomicAdd(&usum[tid], bs[tid]); atomicAdd(&usumsq[tid], bq[tid]); }
}

// ---- Pass 4: BN(up) in place; upsample+BN attn_ch into d_out[4M..8M) -------
__global__ __launch_bounds__(256) void k_final(float* __restrict__ out,
                                               const float* __restrict__ acs,
                                               const float* __restrict__ usum,
                                               const float* __restrict__ usumsq,
                                               const float* __restrict__ asum,
                                               const float* __restrict__ asumsq,
                                               const float* __restrict__ g2d,
                                               const float* __restrict__ b2d,
                                               const float* __restrict__ gat,
                                               const float* __restrict__ bat) {
  const size_t idx = (size_t)blockIdx.x * 256 + threadIdx.x;
  if (idx < 4194304) {                       // up: (B,OC,64,64)
    const int oc = (int)((idx >> 12) & 31);
    const float inv_n = 1.0f / 131072.0f;    // B*64*64
    const float mean = usum[oc] * inv_n;
    const float var  = usumsq[oc] * inv_n - mean * mean;
    const float s = g2d[oc] * rsqrtf(var + 1e-5f);
    const float t = b2d[oc] - mean * s;
    out[idx] = out[idx] * s + t;
  } else {                                   // attn_ch: (B,C,64,64), 2x replicate
    const size_t o = idx - 4194304;
    const int b  = (int)(o >> 17);
    const int ch = (int)((o >> 12) & 31);
    const int y  = (int)((o >> 6) & 63);
    const int xw = (int)(o & 63);
    const float v = acs[(((size_t)(b * 32 + ch)) << 10) + (size_t)(y >> 1) * 32 + (xw >> 1)];
    const float inv_n = 1.0f / 32768.0f;     // replication keeps moments
    const float mean = asum[ch] * inv_n;
    const float var  = asumsq[ch] * inv_n - mean * mean;
    const float s = gat[ch] * rsqrtf(var + 1e-5f);
    const float t = bat[ch] - mean * s;
    out[idx] = v * s + t;
  }
}

extern "C" void kernel_launch(void* const* d_in, const int* in_sizes, int n_in,
                              void* d_out, int out_size, void* d_ws, size_t ws_size,
                              hipStream_t stream) {
  const float* x    = (const float*)d_in[0];
  const float* wct  = (const float*)d_in[1];
  const float* bct  = (const float*)d_in[2];
  const float* g3d  = (const float*)d_in[3];
  const float* b3d  = (const float*)d_in[4];
  const float* g2d  = (const float*)d_in[5];
  const float* b2d  = (const float*)d_in[6];
  const float* gat  = (const float*)d_in[7];
  const float* bat  = (const float*)d_in[8];

  float* out = (float*)d_out;                // 8,388,608 floats
  float* ws  = (float*)d_ws;
  float* ar2 = ws;                           // 262,144
  float* acs = ws + 262144;                  // 1,048,576
  float* acc = ws + 1310720;                 // 192 accumulators
  float* hsum = acc, *hsumsq = acc + 32;
  float* asum = acc + 64, *asumsq = acc + 96;
  float* usum = acc + 128, *usumsq = acc + 160;

  k_init<<<1, 256, 0, stream>>>(acc);
  k_hist<<<1024, 256, 0, stream>>>(x, out, hsum, hsumsq);          // hist staged in d_out
  k_attn<<<128, 256, 0, stream>>>(out, g3d, b3d, hsum, hsumsq, ar2, acs, asum, asumsq);
  k_convt<<<2048, 256, 0, stream>>>(ar2, wct, bct, out, usum, usumsq);
  k_final<<<32768, 256, 0, stream>>>(out, acs, usum, usumsq, asum, asumsq,
                                     g2d, b2d, gat, bat);
}